// TCLWithoutParametersLoss_15917148799617
// MI455X (gfx1250) — compile-verified
//
#include <hip/hip_runtime.h>

typedef __attribute__((ext_vector_type(2))) float v2f;
typedef __attribute__((ext_vector_type(8))) float v8f;

#define DDIM 512
#define NCLS 21
#define MARGIN_F 5.0f

// One wave (32 lanes) computes a 16-row x 32-class (21 real) distance tile via
// V_WMMA_F32_16X16X4_F32, then the margin loss for its 16 rows.
__global__ __launch_bounds__(128) void tcl_main_kernel(
    const float* __restrict__ feats,    // (B, 512)
    const float* __restrict__ centers,  // (21, 512)
    const int*   __restrict__ labels,   // (B,)
    float*       __restrict__ partial,  // (B/16,) per-tile loss sums (pre-scaled by 1/B)
    float invB) {
  const int wave   = threadIdx.x >> 5;
  const int lane   = threadIdx.x & 31;
  const int tile   = blockIdx.x * 4 + wave;   // 16-row tile index
  const int rowB   = tile * 16;

  const int m  = lane & 15;   // row (A) / column (B) index for this lane
  const int kh = lane >> 4;   // K-pair select: 0 -> K={0,1}, 1 -> K={2,3}

  // A fragment source: row (rowB+m), elements k+2*kh, k+2*kh+1
  const float* fRow = feats + (size_t)(rowB + m) * DDIM + 2 * kh;
  // B fragments: tile0 = classes 0..15, tile1 = classes 16..31 (pad >=21 reads class 0)
  const int c0  = m;
  const int c1  = 16 + m;
  const int c1m = (c1 < NCLS) ? c1 : 0;
  const float* b0Row = centers + (size_t)c0  * DDIM + 2 * kh;
  const float* b1Row = centers + (size_t)c1m * DDIM + 2 * kh;

  v8f acc0 = {};
  v8f acc1 = {};
  float f2p = 0.0f, c2p0 = 0.0f, c2p1 = 0.0f;

#pragma unroll 4
  for (int k = 0; k < DDIM; k += 4) {
    v2f a  = *(const v2f*)(fRow  + k);
    v2f b0 = *(const v2f*)(b0Row + k);
    v2f b1 = *(const v2f*)(b1Row + k);
    f2p  = fmaf(a.x,  a.x,  f2p);  f2p  = fmaf(a.y,  a.y,  f2p);
    c2p0 = fmaf(b0.x, b0.x, c2p0); c2p0 = fmaf(b0.y, b0.y, c2p0);
    c2p1 = fmaf(b1.x, b1.x, c2p1); c2p1 = fmaf(b1.y, b1.y, c2p1);
    // D = A(16x4) * B(4x16) + C ; fp32 matrix pipe
    acc0 = __builtin_amdgcn_wmma_f32_16x16x4_f32(false, a, false, b0,
                                                 (short)0, acc0, false, false);
    acc1 = __builtin_amdgcn_wmma_f32_16x16x4_f32(false, a, false, b1,
                                                 (short)0, acc1, false, false);
  }

  // Each lane saw exactly half of its row/column; combine halves (lanes m, m+16).
  float f2full = f2p  + __shfl_xor(f2p,  16, 32);  // lane L holds ||f[row L&15]||^2
  float c2_0   = c2p0 + __shfl_xor(c2p0, 16, 32);  // lane L holds ||c[L&15]||^2
  float c2_1   = c2p1 + __shfl_xor(c2p1, 16, 32);  // lane L holds ||c[16+(L&15)]||^2 (pad: c[0])

  // Preload labels: lane m holds label of row m.
  int labV = labels[rowB + m];

  const float INF = __builtin_inff();
  float wloss = 0.0f;

#pragma unroll
  for (int j = 0; j < 8; ++j) {
    // C/D layout: VGPR j, lane half kh -> row = j + 8*kh, class = (lane&15) [+16 for tile1]
    int   row = j + 8 * kh;
    float f2r = __shfl(f2full, row, 32);
    int   lab = __shfl(labV,   row, 32);

    float d0 = 0.5f * (f2r + c2_0) - acc0[j];   // dist(row, c0)
    float d1 = 0.5f * (f2r + c2_1) - acc1[j];   // dist(row, c1)

    bool is0 = (c0 == lab);
    bool is1 = (c1 == lab);            // never true for padded classes (lab < 21 <= c1)
    bool v1ok = (c1 < NCLS);

    float posc = is0 ? d0 : (is1 ? d1 : -INF);
    float n0   = is0 ? INF : d0;
    float n1   = (v1ok && !is1) ? d1 : INF;
    float negc = fminf(n0, n1);

    // Reduce over the 16 lanes of this half (classes 0..31 for this row).
#pragma unroll
    for (int off = 1; off <= 8; off <<= 1) {
      posc = fmaxf(posc, __shfl_xor(posc, off, 32));
      negc = fminf(negc, __shfl_xor(negc, off, 32));
    }

    float l = posc + MARGIN_F - negc;
    l = l > 0.0f ? l : 0.0f;            // relu; neg=+inf cannot happen (C>1)
    if ((lane & 15) == 0) wloss += l;   // lane 0: rows j (half0); lane 16: rows j+8 (half1)
  }

  float tot = wloss + __shfl_xor(wloss, 16, 32);
  if (lane == 0) partial[tile] = tot * invB;
}

__global__ __launch_bounds__(256) void tcl_reduce_kernel(
    const float* __restrict__ partial, float* __restrict__ out, int n) {
  __shared__ float smem[256];
  float s = 0.0f;
  for (int i = threadIdx.x; i < n; i += 256) s += partial[i];
  smem[threadIdx.x] = s;
  __syncthreads();
  for (int w = 128; w > 0; w >>= 1) {
    if ((int)threadIdx.x < w) smem[threadIdx.x] += smem[threadIdx.x + w];
    __syncthreads();
  }
  if (threadIdx.x == 0) out[0] = smem[0];
}

extern "C" void kernel_launch(void* const* d_in, const int* in_sizes, int n_in,
                              void* d_out, int out_size, void* d_ws, size_t ws_size,
                              hipStream_t stream) {
  const float* feats   = (const float*)d_in[0];
  const float* centers = (const float*)d_in[1];
  const int*   labels  = (const int*)d_in[2];
  float*       out     = (float*)d_out;
  float*       partial = (float*)d_ws;

  const int B     = in_sizes[0] / DDIM;   // 65536
  const int tiles = B / 16;               // 4096
  const float invB = 1.0f / (float)B;

  tcl_main_kernel<<<dim3(tiles / 4), 128, 0, stream>>>(feats, centers, labels,
                                                       partial, invB);
  tcl_reduce_kernel<<<dim3(1), 256, 0, stream>>>(partial, out, tiles);
}